// SolvePoisson_24111946399959
// MI455X (gfx1250) — compile-verified
//
#include <hip/hip_runtime.h>
#include <math.h>

// Problem constants (64x64 lattice, 32 RHS)
#define NN 4096
#define NB 32
#define NITER 200

typedef __attribute__((ext_vector_type(2))) float v2f;
typedef __attribute__((ext_vector_type(8))) float v8f;

// ---------------- workspace layout (float units) ----------------
constexpr int OFF_C     = 0;                      // 4096 conductances
constexpr int OFF_WSUM  = OFF_C + NN;             // 4096 Laplacian diagonal
constexpr int OFF_MEAN  = OFF_WSUM + NN;          // 1 (padded to 64)
constexpr int OFF_S     = OFF_MEAN + 64;          // 32 column sums (padded 64)
constexpr int OFF_ALPHA = OFF_S + 64;             // 32x32
constexpr int OFF_BETA  = OFF_ALPHA + NB * NB;    // 32x32
constexpr int OFF_X     = OFF_BETA + NB * NB;     // NxNB
constexpr int OFF_R     = OFF_X + NN * NB;
constexpr int OFF_P     = OFF_R + NN * NB;
constexpr int OFF_AP    = OFF_P + NN * NB;
constexpr int OFF_PARTA = OFF_AP + NN * NB;       // 128 partial 32x32 grams
constexpr int OFF_PARTB = OFF_PARTA + 128 * NB * NB;
constexpr int OFF_DBL   = OFF_PARTB + 128 * NB * NB;  // even -> 8B aligned
// double region (double units, base = (double*)(ws + OFF_DBL))
constexpr int DOFF_L = 0;   // 32x32 Cholesky factor (persists alpha->beta solve)

__device__ __forceinline__ v8f wmma4(v2f a, v2f b, v8f c) {
  // D(16x16,f32) = A(16x4,f32) * B(4x16,f32) + C  -> v_wmma_f32_16x16x4_f32
  return __builtin_amdgcn_wmma_f32_16x16x4_f32(
      /*neg_a=*/false, a, /*neg_b=*/false, b,
      /*c_mod=*/(short)0, c, /*reuse_a=*/false, /*reuse_b=*/false);
}

// ---------------- setup ----------------
__global__ void sp_setup(const float* __restrict__ lr, float* __restrict__ ws) {
  int i = blockIdx.x * 256 + threadIdx.x;
  int h = i >> 6, w = i & 63;
  float ci = expf(lr[i]);
  float s = 0.f;
  if (w > 0)  s += ci + expf(lr[i - 1]);
  if (w < 63) s += ci + expf(lr[i + 1]);
  if (h > 0)  s += ci + expf(lr[i - 64]);
  if (h < 63) s += ci + expf(lr[i + 64]);
  ws[OFF_C + i] = ci;
  ws[OFF_WSUM + i] = s;
}

__global__ void sp_mean(const float* __restrict__ cur, float* __restrict__ ws) {
  __shared__ double sd[256];
  double a = 0.0;
  for (int idx = threadIdx.x; idx < NN * NB; idx += 256) a += (double)cur[idx];
  sd[threadIdx.x] = a;
  __syncthreads();
  for (int k = 128; k > 0; k >>= 1) {
    if ((int)threadIdx.x < k) sd[threadIdx.x] += sd[threadIdx.x + k];
    __syncthreads();
  }
  if (threadIdx.x == 0) ws[OFF_MEAN] = (float)(sd[0] / (double)(NN * NB));
}

__global__ void sp_init(const float* __restrict__ cur, float* __restrict__ ws) {
  int idx = blockIdx.x * 256 + threadIdx.x;  // idx = node*32 + batch
  int i = idx >> 5, b = idx & 31;
  float v = cur[b * NN + i] - ws[OFF_MEAN];
  ws[OFF_X + idx] = 0.f;
  ws[OFF_R + idx] = v;
  ws[OFF_P + idx] = v;
}

// ---------------- per-iteration kernels ----------------
__global__ void sp_colsum(float* __restrict__ ws) {  // s[b] = sum_i P[i][b]
  __shared__ double sd[256];
  int b = blockIdx.x;
  double a = 0.0;
  for (int i = threadIdx.x; i < NN; i += 256) a += (double)ws[OFF_P + i * NB + b];
  sd[threadIdx.x] = a;
  __syncthreads();
  for (int k = 128; k > 0; k >>= 1) {
    if ((int)threadIdx.x < k) sd[threadIdx.x] += sd[threadIdx.x + k];
    __syncthreads();
  }
  if (threadIdx.x == 0) ws[OFF_S + b] = (float)sd[0];
}

__global__ void sp_apply(float* __restrict__ ws) {  // AP = (L + J) P
  int idx = blockIdx.x * 256 + threadIdx.x;
  int i = idx >> 5, b = idx & 31;
  int h = i >> 6, w = i & 63;
  const float* P = ws + OFF_P;
  float p = P[idx];
  float ci = ws[OFF_C + i];
  float acc = ws[OFF_WSUM + i] * p + ws[OFF_S + b];  // diag + J*P
  if (w > 0)  acc -= (ci + ws[OFF_C + i - 1])  * P[idx - NB];
  if (w < 63) acc -= (ci + ws[OFF_C + i + 1])  * P[idx + NB];
  if (h > 0)  acc -= (ci + ws[OFF_C + i - 64]) * P[idx - 64 * NB];
  if (h < 63) acc -= (ci + ws[OFF_C + i + 64]) * P[idx + 64 * NB];
  ws[OFF_AP + idx] = acc;
}

// Gram partials: G_part = P^T AP, W_part = P^T R   (16 blocks x 8 waves, 32 rows/wave)
__global__ void sp_gram_dual(float* __restrict__ ws) {
  const float* P  = ws + OFF_P;
  const float* AP = ws + OFF_AP;
  const float* R  = ws + OFF_R;
  int lane = threadIdx.x & 31;
  int wave = blockIdx.x * 8 + (threadIdx.x >> 5);
  int rowbase = wave * 32;
  int ml = lane & 15, kh = lane >> 4;
  v8f accG[2][2] = {};
  v8f accW[2][2] = {};
#pragma unroll
  for (int kk = 0; kk < 8; ++kk) {
    int k0 = rowbase + kk * 4;
    v2f a[2], bg[2], bw[2];
#pragma unroll
    for (int mh = 0; mh < 2; ++mh) {  // A = P^T : A[m][k] = P[k0+k][m]
      int m = mh * 16 + ml;
      a[mh].x = P[(k0 + 2 * kh + 0) * NB + m];
      a[mh].y = P[(k0 + 2 * kh + 1) * NB + m];
    }
#pragma unroll
    for (int nh = 0; nh < 2; ++nh) {  // B[k][n]
      int n = nh * 16 + ml;
      bg[nh].x = AP[(k0 + kh) * NB + n];
      bg[nh].y = AP[(k0 + kh + 2) * NB + n];
      bw[nh].x = R[(k0 + kh) * NB + n];
      bw[nh].y = R[(k0 + kh + 2) * NB + n];
    }
#pragma unroll
    for (int mh = 0; mh < 2; ++mh)
#pragma unroll
      for (int nh = 0; nh < 2; ++nh) {
        accG[mh][nh] = wmma4(a[mh], bg[nh], accG[mh][nh]);
        accW[mh][nh] = wmma4(a[mh], bw[nh], accW[mh][nh]);
      }
  }
  float* pg = ws + OFF_PARTA + wave * (NB * NB);
  float* pw = ws + OFF_PARTB + wave * (NB * NB);
#pragma unroll
  for (int mh = 0; mh < 2; ++mh)
#pragma unroll
    for (int nh = 0; nh < 2; ++nh)
#pragma unroll
      for (int r = 0; r < 8; ++r) {
        int M = mh * 16 + kh * 8 + r;  // C layout: VGPR r -> M=r (lo lanes) / r+8 (hi)
        int Nc = nh * 16 + ml;
        pg[M * NB + Nc] = accG[mh][nh][r];
        pw[M * NB + Nc] = accW[mh][nh][r];
      }
}

// W2_part = (AP)^T R
__global__ void sp_gram_single(float* __restrict__ ws) {
  const float* AP = ws + OFF_AP;
  const float* R  = ws + OFF_R;
  int lane = threadIdx.x & 31;
  int wave = blockIdx.x * 8 + (threadIdx.x >> 5);
  int rowbase = wave * 32;
  int ml = lane & 15, kh = lane >> 4;
  v8f acc[2][2] = {};
#pragma unroll
  for (int kk = 0; kk < 8; ++kk) {
    int k0 = rowbase + kk * 4;
    v2f a[2], b[2];
#pragma unroll
    for (int mh = 0; mh < 2; ++mh) {
      int m = mh * 16 + ml;
      a[mh].x = AP[(k0 + 2 * kh + 0) * NB + m];
      a[mh].y = AP[(k0 + 2 * kh + 1) * NB + m];
    }
#pragma unroll
    for (int nh = 0; nh < 2; ++nh) {
      int n = nh * 16 + ml;
      b[nh].x = R[(k0 + kh) * NB + n];
      b[nh].y = R[(k0 + kh + 2) * NB + n];
    }
#pragma unroll
    for (int mh = 0; mh < 2; ++mh)
#pragma unroll
      for (int nh = 0; nh < 2; ++nh) acc[mh][nh] = wmma4(a[mh], b[nh], acc[mh][nh]);
  }
  float* pg = ws + OFF_PARTA + wave * (NB * NB);
#pragma unroll
  for (int mh = 0; mh < 2; ++mh)
#pragma unroll
    for (int nh = 0; nh < 2; ++nh)
#pragma unroll
      for (int r = 0; r < 8; ++r) {
        int M = mh * 16 + kh * 8 + r;
        int Nc = nh * 16 + ml;
        pg[M * NB + Nc] = acc[mh][nh][r];
      }
}

// Fused: combine 128 partial Grams (fp64, in LDS) + Cholesky + solve G*alpha = W1.
// 1 block x 1024 threads; barriers are block-wide (no divergent barriers).
__global__ void sp_reduce_alpha(float* __restrict__ ws) {
  __shared__ double sG[32][33];
  __shared__ double sW[32][32];
  int e = threadIdx.x;
  double g = 0.0, w = 0.0;
  for (int k = 0; k < 128; ++k) {
    g += (double)ws[OFF_PARTA + k * 1024 + e];
    w += (double)ws[OFF_PARTB + k * 1024 + e];
  }
  sG[e >> 5][e & 31] = g;
  sW[e >> 5][e & 31] = w;
  __syncthreads();

  bool active = e < 32;
  int r = e & 31;
  for (int j = 0; j < 32; ++j) {
    if (active && r == j) {
      double d = sG[j][j];
      sG[j][j] = (d > 1e-300 && d < 1e300) ? sqrt(d) : 1.0;
    }
    __syncthreads();
    if (active && r > j) sG[r][j] /= sG[j][j];
    __syncthreads();
    if (active && r > j)
      for (int k = j + 1; k <= r; ++k) sG[r][k] -= sG[r][j] * sG[k][j];
    __syncthreads();
  }
  if (active) {
    double* D = (double*)(ws + OFF_DBL);
    for (int k = 0; k < 32; ++k) D[DOFF_L + r * 32 + k] = sG[r][k];
    // per-thread column solve: L L^T x = W1[:, r]
    double y[32];
    for (int i = 0; i < 32; ++i) {
      double s = sW[i][r];
      for (int k = 0; k < i; ++k) s -= sG[i][k] * y[k];
      y[i] = s / sG[i][i];
    }
    for (int i = 31; i >= 0; --i) {
      double s = y[i];
      for (int k = i + 1; k < 32; ++k) s -= sG[k][i] * y[k];
      y[i] = s / sG[i][i];
    }
    for (int i = 0; i < 32; ++i) {
      float v = (float)y[i];
      if (!(v == v) || fabsf(v) > 1e30f) v = 0.f;  // NaN/Inf guard post-convergence
      ws[OFF_ALPHA + i * 32 + r] = v;
    }
  }
}

// Fused: combine W2 partials + solve with stored L; beta = -G^{-1} (AP)^T R_new.
__global__ void sp_reduce_beta(float* __restrict__ ws) {
  __shared__ double sG[32][33];
  __shared__ double sW[32][32];
  int e = threadIdx.x;
  double g = 0.0;
  for (int k = 0; k < 128; ++k) g += (double)ws[OFF_PARTA + k * 1024 + e];
  sW[e >> 5][e & 31] = g;
  const double* D = (const double*)(ws + OFF_DBL);
  sG[e >> 5][e & 31] = D[DOFF_L + e];
  __syncthreads();

  if (e < 32) {
    int r = e;
    double y[32];
    for (int i = 0; i < 32; ++i) {
      double s = sW[i][r];
      for (int k = 0; k < i; ++k) s -= sG[i][k] * y[k];
      y[i] = s / sG[i][i];
    }
    for (int i = 31; i >= 0; --i) {
      double s = y[i];
      for (int k = i + 1; k < 32; ++k) s -= sG[k][i] * y[k];
      y[i] = s / sG[i][i];
    }
    for (int i = 0; i < 32; ++i) {
      float v = (float)(-y[i]);
      if (!(v == v) || fabsf(v) > 1e30f) v = 0.f;
      ws[OFF_BETA + i * 32 + r] = v;
    }
  }
}

// X += P*alpha ; R -= AP*alpha     (32 blocks x 8 waves, 16 rows/wave)
__global__ void sp_update_XR(float* __restrict__ ws) {
  const v2f* P2  = (const v2f*)(ws + OFF_P);   // K-contiguous rows -> b64 loads
  const v2f* AP2 = (const v2f*)(ws + OFF_AP);
  const float* AL = ws + OFF_ALPHA;
  int lane = threadIdx.x & 31;
  int g = blockIdx.x * 8 + (threadIdx.x >> 5);
  int row0 = g * 16;
  int ml = lane & 15, kh = lane >> 4;
  v8f cx[2] = {};
  v8f cr[2] = {};
#pragma unroll
  for (int kk = 0; kk < 8; ++kk) {
    int k0 = kk * 4;
    int m = row0 + ml;
    v2f ap_ = P2[m * (NB / 2) + kk * 2 + kh];   // P[m][k0+2kh .. +1]
    v2f aa_ = AP2[m * (NB / 2) + kk * 2 + kh];
#pragma unroll
    for (int nh = 0; nh < 2; ++nh) {
      int n = nh * 16 + ml;
      v2f b_;
      b_.x = AL[(k0 + kh) * NB + n];
      b_.y = AL[(k0 + kh + 2) * NB + n];
      cx[nh] = wmma4(ap_, b_, cx[nh]);
      cr[nh] = wmma4(aa_, b_, cr[nh]);
    }
  }
  float* X = ws + OFF_X;
  float* R = ws + OFF_R;
#pragma unroll
  for (int nh = 0; nh < 2; ++nh)
#pragma unroll
    for (int r2 = 0; r2 < 8; ++r2) {
      int M = row0 + kh * 8 + r2;
      int Nc = nh * 16 + ml;
      X[M * NB + Nc] += cx[nh][r2];
      R[M * NB + Nc] -= cr[nh][r2];
    }
}

// P = R + P*beta   (each wave owns its 16 rows: read-before-write is safe)
__global__ void sp_update_P(float* __restrict__ ws) {
  float* P = ws + OFF_P;
  const v2f* P2 = (const v2f*)(ws + OFF_P);
  const float* R = ws + OFF_R;
  const float* BE = ws + OFF_BETA;
  int lane = threadIdx.x & 31;
  int g = blockIdx.x * 8 + (threadIdx.x >> 5);
  int row0 = g * 16;
  int ml = lane & 15, kh = lane >> 4;
  v8f cp[2] = {};
#pragma unroll
  for (int kk = 0; kk < 8; ++kk) {
    int k0 = kk * 4;
    int m = row0 + ml;
    v2f a_ = P2[m * (NB / 2) + kk * 2 + kh];
#pragma unroll
    for (int nh = 0; nh < 2; ++nh) {
      int n = nh * 16 + ml;
      v2f b_;
      b_.x = BE[(k0 + kh) * NB + n];
      b_.y = BE[(k0 + kh + 2) * NB + n];
      cp[nh] = wmma4(a_, b_, cp[nh]);
    }
  }
#pragma unroll
  for (int nh = 0; nh < 2; ++nh)
#pragma unroll
    for (int r2 = 0; r2 < 8; ++r2) {
      int M = row0 + kh * 8 + r2;
      int Nc = nh * 16 + ml;
      P[M * NB + Nc] = R[M * NB + Nc] + cp[nh][r2];
    }
}

__global__ void sp_out(const float* __restrict__ ws, float* __restrict__ out) {
  int idx = blockIdx.x * 256 + threadIdx.x;  // node-major -> batch-major
  int i = idx >> 5, b = idx & 31;
  out[b * NN + i] = ws[OFF_X + idx];
}

// ---------------- driver ----------------
extern "C" void kernel_launch(void* const* d_in, const int* in_sizes, int n_in,
                              void* d_out, int out_size, void* d_ws, size_t ws_size,
                              hipStream_t stream) {
  (void)in_sizes; (void)n_in; (void)out_size; (void)ws_size;
  const float* lr  = (const float*)d_in[0];   // log_resistances [4096]
  const float* cur = (const float*)d_in[1];   // input_currents  [32*4096]
  float* out = (float*)d_out;
  float* ws  = (float*)d_ws;

  sp_setup<<<NN / 256, 256, 0, stream>>>(lr, ws);
  sp_mean<<<1, 256, 0, stream>>>(cur, ws);
  sp_init<<<(NN * NB) / 256, 256, 0, stream>>>(cur, ws);

  for (int it = 0; it < NITER; ++it) {
    sp_colsum<<<NB, 256, 0, stream>>>(ws);
    sp_apply<<<(NN * NB) / 256, 256, 0, stream>>>(ws);
    sp_gram_dual<<<16, 256, 0, stream>>>(ws);
    sp_reduce_alpha<<<1, 1024, 0, stream>>>(ws);
    sp_update_XR<<<32, 256, 0, stream>>>(ws);
    sp_gram_single<<<16, 256, 0, stream>>>(ws);
    sp_reduce_beta<<<1, 1024, 0, stream>>>(ws);
    sp_update_P<<<32, 256, 0, stream>>>(ws);
  }

  sp_out<<<(NN * NB) / 256, 256, 0, stream>>>(ws, out);
}